// DAGMM_42958262894931
// MI455X (gfx1250) — compile-verified
//
#include <hip/hip_runtime.h>
#include <math.h>

#define NS 400000
#define XD 128

typedef __attribute__((ext_vector_type(2))) float v2f;
typedef __attribute__((ext_vector_type(8))) float v8f;

struct Weights {
    const float *x1;
    const float *enc_w1, *enc_b1, *enc_w2, *enc_b2, *enc_w3, *enc_b3;
    const float *dec_w1, *dec_b1, *dec_w2, *dec_b2, *dec_w3, *dec_b3;
    const float *est_w1, *est_b1, *est_w2, *est_b2;
};

// small-weight LDS offsets
#define SM_B1   0      // 16
#define SM_W2   16     // 128 (16x8)
#define SM_B2   144    // 8
#define SM_W3   152    // 8
#define SM_B3   160    // 1
#define SM_DW1  161    // 8
#define SM_DB1  169    // 8
#define SM_DW2  177    // 128 (8x16)
#define SM_DB2  305    // 16
#define SM_DB3  321    // 128
#define SM_EW1  449    // 24 (3x8)
#define SM_EB1  473    // 8
#define SM_EW2  481    // 32 (8x4)
#define SM_EB2  513    // 4
#define SM_TOT  517

// ws float layout: [0,3N) z-columns | gAcc 41 | gPar 53 | gE 1
// gAcc: [0..3] gamma_sum, [4..15] gamma*z, [16..39] gamma*z*z upper-tri (4x6), [40] recon_sum
// gPar: [0..11] mu, [12..47] sigma_inv (4x9), [48..51] Ck, [52] loss1+loss3

__global__ void k0_zero(float* p, int n) {
    int i = threadIdx.x;
    if (i < n) p[i] = 0.0f;
}

__global__ __launch_bounds__(128) void k1_forward(Weights W, float* zc, float* gAcc) {
    __shared__ float s_w1[2048];          // enc_w1 128x16
    __shared__ float s_dw3[2048];         // dec_w3 16x128
    __shared__ float s_small[SM_TOT];
    __shared__ float s_h[4 * 32 * 17];    // per-wave transpose buffer (padded)
    __shared__ float s_acc[41];

    const int tid = threadIdx.x;
    for (int i = tid; i < 2048; i += 128) s_w1[i] = W.enc_w1[i];
    for (int i = tid; i < 2048; i += 128) s_dw3[i] = W.dec_w3[i];
    if (tid < 16)  s_small[SM_B1  + tid] = W.enc_b1[tid];
    if (tid < 128) s_small[SM_W2  + tid] = W.enc_w2[tid];
    if (tid < 8)   s_small[SM_B2  + tid] = W.enc_b2[tid];
    if (tid < 8)   s_small[SM_W3  + tid] = W.enc_w3[tid];
    if (tid < 1)   s_small[SM_B3  + tid] = W.enc_b3[tid];
    if (tid < 8)   s_small[SM_DW1 + tid] = W.dec_w1[tid];
    if (tid < 8)   s_small[SM_DB1 + tid] = W.dec_b1[tid];
    if (tid < 128) s_small[SM_DW2 + tid] = W.dec_w2[tid];
    if (tid < 16)  s_small[SM_DB2 + tid] = W.dec_b2[tid];
    if (tid < 128) s_small[SM_DB3 + tid] = W.dec_b3[tid];
    if (tid < 24)  s_small[SM_EW1 + tid] = W.est_w1[tid];
    if (tid < 8)   s_small[SM_EB1 + tid] = W.est_b1[tid];
    if (tid < 32)  s_small[SM_EW2 + tid] = W.est_w2[tid];
    if (tid < 4)   s_small[SM_EB2 + tid] = W.est_b2[tid];
    if (tid < 41)  s_acc[tid] = 0.0f;
    __syncthreads();

    const int w    = tid >> 5;
    const int lane = tid & 31;
    const int m    = lane & 15;            // row (A) / col (B/C)
    const int koff = (lane >> 4) << 1;     // 0 or 2
    const long sbase = (long)blockIdx.x * 128 + (long)w * 32;

    // ---- enc layer 1 via V_WMMA_F32_16X16X4_F32, K chained 0..127 ----
    const float* rowA0 = W.x1 + (sbase + m) * XD;
    const float* rowA1 = W.x1 + (sbase + 16 + m) * XD;
    v8f acc0 = {}, acc1 = {};
#pragma unroll
    for (int kk = 0; kk < 32; ++kk) {
        const int k = kk * 4;
        v2f a0, a1, b;
        a0.x = rowA0[k + koff];     a0.y = rowA0[k + koff + 1];
        a1.x = rowA1[k + koff];     a1.y = rowA1[k + koff + 1];
        b.x  = s_w1[(k + koff) * 16 + m];
        b.y  = s_w1[(k + koff + 1) * 16 + m];
        acc0 = __builtin_amdgcn_wmma_f32_16x16x4_f32(false, a0, false, b, (short)0, acc0, false, false);
        acc1 = __builtin_amdgcn_wmma_f32_16x16x4_f32(false, a1, false, b, (short)0, acc1, false, false);
    }

    // bias + tanh, transpose D (lane=neuron-column) -> LDS (row=sample)
    const float b1v = s_small[SM_B1 + m];
    float* hb = s_h + w * (32 * 17);
#pragma unroll
    for (int r = 0; r < 8; ++r) {
        const int s0 = ((lane >> 4) << 3) + r;     // sample-in-tile 0..15
        hb[s0 * 17 + m]        = tanhf(acc0[r] + b1v);
        hb[(16 + s0) * 17 + m] = tanhf(acc1[r] + b1v);
    }
    __syncthreads();

    // ---- per-lane = per-sample downstream ----
    float h1[16];
#pragma unroll
    for (int i = 0; i < 16; ++i) h1[i] = hb[lane * 17 + i];

    float h2[8];
#pragma unroll
    for (int j = 0; j < 8; ++j) {
        float s = s_small[SM_B2 + j];
#pragma unroll
        for (int i = 0; i < 16; ++i) s = fmaf(h1[i], s_small[SM_W2 + i * 8 + j], s);
        h2[j] = tanhf(s);
    }
    float z1v = s_small[SM_B3];
#pragma unroll
    for (int i = 0; i < 8; ++i) z1v = fmaf(h2[i], s_small[SM_W3 + i], z1v);
    z1v = tanhf(z1v);

    float d1[8];
#pragma unroll
    for (int j = 0; j < 8; ++j)
        d1[j] = tanhf(fmaf(z1v, s_small[SM_DW1 + j], s_small[SM_DB1 + j]));

    float d2[16];
#pragma unroll
    for (int j = 0; j < 16; ++j) {
        float s = s_small[SM_DB2 + j];
#pragma unroll
        for (int i = 0; i < 8; ++i) s = fmaf(d1[i], s_small[SM_DW2 + i * 16 + j], s);
        d2[j] = tanhf(s);
    }

    // dec layer 3 + distances, streaming over xdim
    const float* xrow = W.x1 + (sbase + lane) * XD;
    float dot = 0.f, nx1 = 0.f, nx2 = 0.f, rec = 0.f;
#pragma unroll 4
    for (int j = 0; j < XD; ++j) {
        float s = s_small[SM_DB3 + j];
#pragma unroll
        for (int i = 0; i < 16; ++i) s = fmaf(d2[i], s_dw3[i * XD + j], s);
        const float xv = xrow[j];
        dot = fmaf(xv, s, dot);
        nx1 = fmaf(xv, xv, nx1);
        nx2 = fmaf(s, s, nx2);
        const float df = xv - s;
        rec = fmaf(df, df, rec);
    }
    const float cosv = dot / sqrtf(nx1 * nx2);
    const float eucv = sqrtf(rec);

    // estimator + softmax
    float e1[8];
#pragma unroll
    for (int j = 0; j < 8; ++j) {
        float s = s_small[SM_EB1 + j];
        s = fmaf(z1v,  s_small[SM_EW1 + j],      s);
        s = fmaf(cosv, s_small[SM_EW1 + 8 + j],  s);
        s = fmaf(eucv, s_small[SM_EW1 + 16 + j], s);
        e1[j] = tanhf(s);
    }
    float lg[4];
#pragma unroll
    for (int k = 0; k < 4; ++k) {
        float s = s_small[SM_EB2 + k];
#pragma unroll
        for (int i = 0; i < 8; ++i) s = fmaf(e1[i], s_small[SM_EW2 + i * 4 + k], s);
        lg[k] = s;
    }
    float mx = fmaxf(fmaxf(lg[0], lg[1]), fmaxf(lg[2], lg[3]));
    float ex[4], esum = 0.f;
#pragma unroll
    for (int k = 0; k < 4; ++k) { ex[k] = expf(lg[k] - mx); esum += ex[k]; }
    const float rinv = 1.0f / esum;

    const long n = sbase + lane;
    zc[n]            = z1v;
    zc[NS + n]       = cosv;
    zc[2L * NS + n]  = eucv;

    const float zv[3] = { z1v, cosv, eucv };
#pragma unroll
    for (int k = 0; k < 4; ++k) {
        const float g = ex[k] * rinv;
        atomicAdd(&s_acc[k], g);
#pragma unroll
        for (int d = 0; d < 3; ++d) atomicAdd(&s_acc[4 + k * 3 + d], g * zv[d]);
        int c = 0;
#pragma unroll
        for (int d = 0; d < 3; ++d)
#pragma unroll
            for (int e = d; e < 3; ++e)
                atomicAdd(&s_acc[16 + k * 6 + (c++)], g * zv[d] * zv[e]);
    }
    atomicAdd(&s_acc[40], rec);
    __syncthreads();
    if (tid < 41) atomicAdd(&gAcc[tid], s_acc[tid]);
}

__global__ void k2_finalize(const float* gAcc, float* gPar) {
    if (threadIdx.x != 0) return;
    const float Nf = (float)NS;
    const float TP3 = 248.05021344239853f;   // (2*pi)^3
    float extra = 0.0f;
    for (int k = 0; k < 4; ++k) {
        const float g = gAcc[k];
        const float m0 = gAcc[4 + k * 3 + 0] / g;
        const float m1 = gAcc[4 + k * 3 + 1] / g;
        const float m2 = gAcc[4 + k * 3 + 2] / g;
        const float* U = gAcc + 16 + k * 6;
        const float S00 = U[0] / g - m0 * m0, S01 = U[1] / g - m0 * m1, S02 = U[2] / g - m0 * m2;
        const float S11 = U[3] / g - m1 * m1, S12 = U[4] / g - m1 * m2, S22 = U[5] / g - m2 * m2;
        const float c00 = S11 * S22 - S12 * S12;
        const float c01 = S02 * S12 - S01 * S22;
        const float c02 = S01 * S12 - S02 * S11;
        const float det = S00 * c00 + S01 * c01 + S02 * c02;
        const float iv  = 1.0f / det;
        gPar[k * 3 + 0] = m0; gPar[k * 3 + 1] = m1; gPar[k * 3 + 2] = m2;
        float* I = gPar + 12 + k * 9;
        I[0] = c00 * iv; I[1] = c01 * iv; I[2] = c02 * iv;
        I[3] = c01 * iv; I[4] = (S00 * S22 - S02 * S02) * iv; I[5] = (S02 * S01 - S00 * S12) * iv;
        I[6] = c02 * iv; I[7] = I[5];     I[8] = (S00 * S11 - S01 * S01) * iv;
        gPar[48 + k] = logf(g / Nf) - 0.5f * logf(TP3 * det);
        extra += 1e-4f * (1.0f / S00 + 1.0f / S11 + 1.0f / S22);
    }
    gPar[52] = gAcc[40] / Nf + extra;   // loss1 + loss3
}

__global__ __launch_bounds__(256) void k3_energy(const float* zc, const float* gPar,
                                                 float* out, float* gE) {
    __shared__ float s_sum;
    __shared__ float s_mu[12], s_iv[36], s_ck[4];
    const int tid = threadIdx.x;
    if (tid == 0) s_sum = 0.0f;
    if (tid < 12) s_mu[tid] = gPar[tid];
    if (tid < 36) s_iv[tid] = gPar[12 + tid];
    if (tid < 4)  s_ck[tid] = gPar[48 + tid];
    __syncthreads();
    const long n = (long)blockIdx.x * 256 + tid;
    float e = 0.0f;
    if (n < NS) {
        const float z0 = zc[n], z1 = zc[NS + n], z2 = zc[2L * NS + n];
        float acc = 0.0f;
#pragma unroll
        for (int k = 0; k < 4; ++k) {
            const float a = z0 - s_mu[k * 3 + 0];
            const float b = z1 - s_mu[k * 3 + 1];
            const float c = z2 - s_mu[k * 3 + 2];
            const float* I = s_iv + k * 9;
            const float q = a * (I[0] * a + I[1] * b + I[2] * c)
                          + b * (I[3] * a + I[4] * b + I[5] * c)
                          + c * (I[6] * a + I[7] * b + I[8] * c);
            acc += s_ck[k] - 0.5f * q;
        }
        e = -acc;
        out[n] = e;
    }
    atomicAdd(&s_sum, e);
    __syncthreads();
    if (tid == 0) atomicAdd(gE, s_sum);
}

__global__ void k4_loss(const float* gPar, const float* gE, float* out) {
    out[NS] = gPar[52] + 0.01f * (gE[0] / (float)NS);
}

extern "C" void kernel_launch(void* const* d_in, const int* in_sizes, int n_in,
                              void* d_out, int out_size, void* d_ws, size_t ws_size,
                              hipStream_t stream) {
    Weights W;
    W.x1     = (const float*)d_in[0];
    W.enc_w1 = (const float*)d_in[1];  W.enc_b1 = (const float*)d_in[2];
    W.enc_w2 = (const float*)d_in[3];  W.enc_b2 = (const float*)d_in[4];
    W.enc_w3 = (const float*)d_in[5];  W.enc_b3 = (const float*)d_in[6];
    W.dec_w1 = (const float*)d_in[7];  W.dec_b1 = (const float*)d_in[8];
    W.dec_w2 = (const float*)d_in[9];  W.dec_b2 = (const float*)d_in[10];
    W.dec_w3 = (const float*)d_in[11]; W.dec_b3 = (const float*)d_in[12];
    W.est_w1 = (const float*)d_in[13]; W.est_b1 = (const float*)d_in[14];
    W.est_w2 = (const float*)d_in[15]; W.est_b2 = (const float*)d_in[16];

    float* ws   = (float*)d_ws;
    float* zc   = ws;                       // 3*NS floats
    float* gAcc = ws + 3L * NS;             // 41
    float* gPar = gAcc + 41;                // 53
    float* gE   = gPar + 53;                // 1

    k0_zero<<<1, 128, 0, stream>>>(gAcc, 95);
    k1_forward<<<3125, 128, 0, stream>>>(W, zc, gAcc);     // 3125*128 == 400000 exactly
    k2_finalize<<<1, 32, 0, stream>>>(gAcc, gPar);
    k3_energy<<<(NS + 255) / 256, 256, 0, stream>>>(zc, gPar, (float*)d_out, gE);
    k4_loss<<<1, 1, 0, stream>>>(gPar, gE, (float*)d_out);
}